// AdvancedGraphSineKANEncoder_11072425689328
// MI455X (gfx1250) — compile-verified
//
#include <hip/hip_runtime.h>
#include <hip/hip_bf16.h>
#include <math.h>

typedef float v2f __attribute__((ext_vector_type(2)));
typedef float v8f __attribute__((ext_vector_type(8)));

#define N_NODES 20000
#define N_EDGES 30000
#define DM      256
#define EDGE_DIM 32
#define PI_F 3.14159265358979323846f

#define WMMA_F32(A, B, C) \
    __builtin_amdgcn_wmma_f32_16x16x4_f32(false, (A), false, (B), (short)0, (C), false, false)

// ---------------- helpers ----------------
__device__ __forceinline__ float wave_sum(float v) {
    v += __shfl_xor(v, 16, 32);
    v += __shfl_xor(v, 8, 32);
    v += __shfl_xor(v, 4, 32);
    v += __shfl_xor(v, 2, 32);
    v += __shfl_xor(v, 1, 32);
    return v;
}

// paired-K LDS index for sin chunks: [pair][col=edge*2+(kl&1)] with parity swizzle
__device__ __forceinline__ int sidx(int kl, int e) {
    int p = kl >> 1;
    int col = (e * 2 + (kl & 1)) ^ ((p & 1) << 5);
    return p * 64 + col;
}

__global__ __launch_bounds__(256) void fill_kernel(float* __restrict__ p, float v, int n) {
    int i = blockIdx.x * 256 + threadIdx.x;
    if (i < n) p[i] = v;
}

// amp [L,O,I,G] -> paired ampT: t[l][k>>1][o][k&1]  (b64-loadable B fragments)
// also packs proj_W [K,O] -> tpw[k>>1][o][k&1]
__global__ __launch_bounds__(256) void transpose_amp(const float* __restrict__ msg_amp,
                                                     const float* __restrict__ upd_amp,
                                                     const float* __restrict__ pW,
                                                     float* __restrict__ tmsg,
                                                     float* __restrict__ tupd,
                                                     float* __restrict__ tpw) {
    const int MSGN = 3 * 2720 * 256;
    const int UPDN = 3 * 2560 * 256;
    const int PWN  = 256 * 256;
    int j = blockIdx.x * 256 + threadIdx.x;
    if (j < MSGN) {
        int o = j & 255;
        int rest = j >> 8;
        int k = rest % 2720, l = rest / 2720;
        tmsg[(size_t)l * 2720 * 256 + (size_t)(k >> 1) * 512 + o * 2 + (k & 1)] =
            msg_amp[((size_t)l * 256 + o) * 2720 + k];
    }
    if (j < UPDN) {
        int o = j & 255;
        int rest = j >> 8;
        int k = rest % 2560, l = rest / 2560;
        tupd[(size_t)l * 2560 * 256 + (size_t)(k >> 1) * 512 + o * 2 + (k & 1)] =
            upd_amp[((size_t)l * 256 + o) * 2560 + k];
    }
    if (j < PWN) {
        int o = j & 255;
        int k = j >> 8;
        tpw[(size_t)(k >> 1) * 512 + o * 2 + (k & 1)] = pW[k * 256 + o];
    }
}

// h = x@node_W + node_b + idx_emb[n%200] + rwse@rwse_W + rwse_b
__global__ __launch_bounds__(256) void enc_kernel(const float* __restrict__ x,
                                                  const float* __restrict__ rwse,
                                                  const float* __restrict__ nW,
                                                  const float* __restrict__ nb,
                                                  const float* __restrict__ idxw,
                                                  const float* __restrict__ rW,
                                                  const float* __restrict__ rb,
                                                  float* __restrict__ h) {
    const int n = blockIdx.x;
    const int o = threadIdx.x;
    float acc = nb[o] + rb[o] + idxw[(n % 200) * 256 + o];
#pragma unroll 8
    for (int k = 0; k < 64; ++k) acc = fmaf(x[n * 64 + k], nW[k * 256 + o], acc);
#pragma unroll
    for (int k = 0; k < 16; ++k) acc = fmaf(rwse[n * 16 + k], rW[k * 256 + o], acc);
    h[(size_t)n * 256 + o] = acc;
}

// ---------------- message sinekan + scatter-add (I=544, K=2720), M-tile=32 ----------------
__global__ __launch_bounds__(256) void msg_kernel(const float* __restrict__ h,
                                                  const int* __restrict__ ei,
                                                  const float* __restrict__ eattr,
                                                  const float* __restrict__ ampT,
                                                  const float* __restrict__ bias,
                                                  float* __restrict__ aggr) {
    __shared__ float xcat[32 * 545];   // 544 padded to 545
    __shared__ float sld[272 * 32];    // paired sin chunk
    __shared__ int dstid[32];
    const int tid = threadIdx.x;
    const int e0 = blockIdx.x << 5;
    if (tid < 32) {
        int ge = e0 + tid;
        dstid[tid] = (ge < N_EDGES) ? ei[N_EDGES + ge] : -1;
    }
    for (int j = tid; j < 32 * 544; j += 256) {
        int e = j / 544, c = j - e * 544;
        int ge = e0 + e;
        float v = 0.0f;
        if (ge < N_EDGES) {
            if (c < 256)      v = h[(size_t)ei[N_EDGES + ge] * 256 + c];      // h[dst]
            else if (c < 512) v = h[(size_t)ei[ge] * 256 + (c - 256)];        // h[src]
            else              v = eattr[(size_t)ge * EDGE_DIM + (c - 512)];
        }
        xcat[e * 545 + c] = v;
    }
    const int wave = tid >> 5, lane = tid & 31, half = lane >> 4, l15 = lane & 15;
    const int ct0 = wave * 2, ct1 = wave * 2 + 1;
    const int a0off = half * 64 + ((2 * l15) ^ (half << 5));        // pl&1 == half
    const int a1off = half * 64 + ((32 + 2 * l15) ^ (half << 5));
    v8f acc00 = {}, acc01 = {}, acc10 = {}, acc11 = {};
    for (int ch = 0; ch < 10; ++ch) {
        const int k0 = ch * 272;
        __syncthreads();
        for (int j = tid; j < 272 * 32; j += 256) {
            int e = j & 31, kl = j >> 5, k = k0 + kl;
            int i = k / 5, g = k - i * 5;
            float f = (float)(g + 1) * (1.0f / 6.0f);
            float ph = f + (PI_F / 543.0f) * (float)i;
            sld[sidx(kl, e)] = __sinf(fmaf(xcat[e * 545 + i], f, ph));
        }
        __syncthreads();
        const float* bp = ampT + (size_t)((k0 >> 1) + half) * 512;
#pragma unroll 4
        for (int s = 0; s < 68; ++s) {
            v2f a0 = *(const v2f*)&sld[s * 128 + a0off];
            v2f a1 = *(const v2f*)&sld[s * 128 + a1off];
            v2f b0 = *(const v2f*)(bp + (ct0 * 16 + l15) * 2);
            v2f b1 = *(const v2f*)(bp + (ct1 * 16 + l15) * 2);
            __builtin_prefetch(bp + 1024, 0, 0);
            acc00 = WMMA_F32(a0, b0, acc00);
            acc10 = WMMA_F32(a1, b0, acc10);
            acc01 = WMMA_F32(a0, b1, acc01);
            acc11 = WMMA_F32(a1, b1, acc11);
            bp += 1024;
        }
    }
    const float bv0 = bias[ct0 * 16 + l15];
    const float bv1 = bias[ct1 * 16 + l15];
#pragma unroll
    for (int r = 0; r < 8; ++r) {
        int m = r + half * 8;
        int d0 = dstid[m];
        int d1 = dstid[16 + m];
        if (d0 >= 0) {
            size_t base = (size_t)d0 * 256;
            __hip_atomic_fetch_add(&aggr[base + ct0 * 16 + l15], acc00[r] + bv0,
                                   __ATOMIC_RELAXED, __HIP_MEMORY_SCOPE_AGENT);
            __hip_atomic_fetch_add(&aggr[base + ct1 * 16 + l15], acc01[r] + bv1,
                                   __ATOMIC_RELAXED, __HIP_MEMORY_SCOPE_AGENT);
        }
        if (d1 >= 0) {
            size_t base = (size_t)d1 * 256;
            __hip_atomic_fetch_add(&aggr[base + ct0 * 16 + l15], acc10[r] + bv0,
                                   __ATOMIC_RELAXED, __HIP_MEMORY_SCOPE_AGENT);
            __hip_atomic_fetch_add(&aggr[base + ct1 * 16 + l15], acc11[r] + bv1,
                                   __ATOMIC_RELAXED, __HIP_MEMORY_SCOPE_AGENT);
        }
    }
}

// ---------------- update sinekan (I=512, K=2560) + two LayerNorms, M-tile=32 ----------------
__global__ __launch_bounds__(256) void upd_kernel(float* __restrict__ h,
                                                  const float* __restrict__ aggr,
                                                  const float* __restrict__ ampT,
                                                  const float* __restrict__ bias,
                                                  const float* __restrict__ cg,
                                                  const float* __restrict__ cb,
                                                  const float* __restrict__ ng,
                                                  const float* __restrict__ nb) {
    __shared__ float xcat[32 * 513];
    __shared__ float buf[320 * 32];    // paired sin chunk during K loop; hnew (32x256) afterwards
    const int tid = threadIdx.x;
    const int n0 = blockIdx.x << 5;
    for (int j = tid; j < 32 * 512; j += 256) {
        int e = j >> 9, c = j & 511;
        float v = (c < 256) ? h[(size_t)(n0 + e) * 256 + c]
                            : aggr[(size_t)(n0 + e) * 256 + (c - 256)];
        xcat[e * 513 + c] = v;
    }
    const int wave = tid >> 5, lane = tid & 31, half = lane >> 4, l15 = lane & 15;
    const int ct0 = wave * 2, ct1 = wave * 2 + 1;
    const int a0off = half * 64 + ((2 * l15) ^ (half << 5));
    const int a1off = half * 64 + ((32 + 2 * l15) ^ (half << 5));
    v8f acc00 = {}, acc01 = {}, acc10 = {}, acc11 = {};
    for (int ch = 0; ch < 8; ++ch) {
        const int k0 = ch * 320;
        __syncthreads();
        for (int j = tid; j < 320 * 32; j += 256) {
            int e = j & 31, kl = j >> 5, k = k0 + kl;
            int i = k / 5, g = k - i * 5;
            float f = (float)(g + 1) * (1.0f / 6.0f);
            float ph = f + (PI_F / 511.0f) * (float)i;
            buf[sidx(kl, e)] = __sinf(fmaf(xcat[e * 513 + i], f, ph));
        }
        __syncthreads();
        const float* bp = ampT + (size_t)((k0 >> 1) + half) * 512;
#pragma unroll 4
        for (int s = 0; s < 80; ++s) {
            v2f a0 = *(const v2f*)&buf[s * 128 + a0off];
            v2f a1 = *(const v2f*)&buf[s * 128 + a1off];
            v2f b0 = *(const v2f*)(bp + (ct0 * 16 + l15) * 2);
            v2f b1 = *(const v2f*)(bp + (ct1 * 16 + l15) * 2);
            __builtin_prefetch(bp + 1024, 0, 0);
            acc00 = WMMA_F32(a0, b0, acc00);
            acc10 = WMMA_F32(a1, b0, acc10);
            acc01 = WMMA_F32(a0, b1, acc01);
            acc11 = WMMA_F32(a1, b1, acc11);
            bp += 1024;
        }
    }
    __syncthreads();
    float* hnew = buf;   // reuse sin-chunk LDS as 32x256 staging
    const float bv0 = bias[ct0 * 16 + l15];
    const float bv1 = bias[ct1 * 16 + l15];
#pragma unroll
    for (int r = 0; r < 8; ++r) {
        int m = r + half * 8;
        hnew[m * 256 + ct0 * 16 + l15]        = acc00[r] + bv0;
        hnew[m * 256 + ct1 * 16 + l15]        = acc01[r] + bv1;
        hnew[(16 + m) * 256 + ct0 * 16 + l15] = acc10[r] + bv0;
        hnew[(16 + m) * 256 + ct1 * 16 + l15] = acc11[r] + bv1;
    }
    __syncthreads();
    // per-row LN(conv) -> residual add -> LN(norm), in-place update of h
    for (int row = wave; row < 32; row += 8) {
        float v[8];
        float s1 = 0.f;
#pragma unroll
        for (int q = 0; q < 8; ++q) { v[q] = hnew[row * 256 + lane + 32 * q]; s1 += v[q]; }
        s1 = wave_sum(s1);
        float mu = s1 * (1.0f / 256.0f);
        float s2 = 0.f;
#pragma unroll
        for (int q = 0; q < 8; ++q) { float d = v[q] - mu; s2 += d * d; }
        s2 = wave_sum(s2);
        float rs = rsqrtf(s2 * (1.0f / 256.0f) + 1e-5f);
        float t[8]; float s3 = 0.f;
#pragma unroll
        for (int q = 0; q < 8; ++q) {
            int o = lane + 32 * q;
            float y = (v[q] - mu) * rs * cg[o] + cb[o];
            t[q] = h[(size_t)(n0 + row) * 256 + o] + y;
            s3 += t[q];
        }
        s3 = wave_sum(s3);
        float mu2 = s3 * (1.0f / 256.0f);
        float s4 = 0.f;
#pragma unroll
        for (int q = 0; q < 8; ++q) { float d = t[q] - mu2; s4 += d * d; }
        s4 = wave_sum(s4);
        float rs2 = rsqrtf(s4 * (1.0f / 256.0f) + 1e-5f);
#pragma unroll
        for (int q = 0; q < 8; ++q) {
            int o = lane + 32 * q;
            h[(size_t)(n0 + row) * 256 + o] = (t[q] - mu2) * rs2 * ng[o] + nb[o];
        }
    }
}

// ---------------- final projection out = h @ proj_W + proj_b, M-tile=32 ----------------
__global__ __launch_bounds__(256) void proj_kernel(const float* __restrict__ h,
                                                   const float* __restrict__ Wp,   // paired tpw
                                                   const float* __restrict__ b,
                                                   float* __restrict__ out) {
    __shared__ float ht[32 * 258];     // even stride -> b64 A fragments
    const int tid = threadIdx.x;
    const int n0 = blockIdx.x << 5;
    for (int j = tid; j < 32 * 256; j += 256) {
        int e = j >> 8, c = j & 255;
        ht[e * 258 + c] = h[(size_t)(n0 + e) * 256 + c];
    }
    __syncthreads();
    const int wave = tid >> 5, lane = tid & 31, half = lane >> 4, l15 = lane & 15;
    const int ct0 = wave * 2, ct1 = wave * 2 + 1;
    v8f acc00 = {}, acc01 = {}, acc10 = {}, acc11 = {};
    const float* bp = Wp + (size_t)half * 512;
#pragma unroll 8
    for (int s = 0; s < 64; ++s) {
        int k = s * 4 + half * 2;
        v2f a0 = *(const v2f*)&ht[l15 * 258 + k];
        v2f a1 = *(const v2f*)&ht[(16 + l15) * 258 + k];
        v2f b0 = *(const v2f*)(bp + (ct0 * 16 + l15) * 2);
        v2f b1 = *(const v2f*)(bp + (ct1 * 16 + l15) * 2);
        acc00 = WMMA_F32(a0, b0, acc00);
        acc10 = WMMA_F32(a1, b0, acc10);
        acc01 = WMMA_F32(a0, b1, acc01);
        acc11 = WMMA_F32(a1, b1, acc11);
        bp += 1024;
    }
    const float bv0 = b[ct0 * 16 + l15];
    const float bv1 = b[ct1 * 16 + l15];
#pragma unroll
    for (int r = 0; r < 8; ++r) {
        int m = r + half * 8;
        size_t base0 = (size_t)(n0 + m) * 256;        // dense-batch layout == node layout
        size_t base1 = (size_t)(n0 + 16 + m) * 256;
        out[base0 + ct0 * 16 + l15] = acc00[r] + bv0;
        out[base0 + ct1 * 16 + l15] = acc01[r] + bv1;
        out[base1 + ct0 * 16 + l15] = acc10[r] + bv0;
        out[base1 + ct1 * 16 + l15] = acc11[r] + bv1;
    }
}

// ---------------- launcher ----------------
extern "C" void kernel_launch(void* const* d_in, const int* in_sizes, int n_in,
                              void* d_out, int out_size, void* d_ws, size_t ws_size,
                              hipStream_t stream) {
    const float* x       = (const float*)d_in[0];
    const int*   ei      = (const int*)d_in[1];
    const float* eattr   = (const float*)d_in[3];
    const float* rwse    = (const float*)d_in[4];
    const float* nW      = (const float*)d_in[5];
    const float* nb      = (const float*)d_in[6];
    const float* idxw    = (const float*)d_in[7];
    const float* rW      = (const float*)d_in[8];
    const float* rb      = (const float*)d_in[9];
    const float* msg_amp = (const float*)d_in[10];
    const float* msg_b   = (const float*)d_in[11];
    const float* upd_amp = (const float*)d_in[12];
    const float* upd_b   = (const float*)d_in[13];
    const float* cg      = (const float*)d_in[14];
    const float* cb      = (const float*)d_in[15];
    const float* ng      = (const float*)d_in[16];
    const float* nb2     = (const float*)d_in[17];
    const float* pW      = (const float*)d_in[18];
    const float* pb      = (const float*)d_in[19];

    float* ws   = (float*)d_ws;
    float* h    = ws;                     // 5,120,000
    float* aggr = ws + 5120000;           // 5,120,000
    float* tmsg = ws + 10240000;          // 3*2720*256 = 2,088,960
    float* tupd = ws + 12328960;          // 3*2560*256 = 1,966,080
    float* tpw  = ws + 14295040;          // 256*256    =    65,536

    transpose_amp<<<(3 * 2720 * 256 + 255) / 256, 256, 0, stream>>>(msg_amp, upd_amp, pW,
                                                                    tmsg, tupd, tpw);
    enc_kernel<<<N_NODES, 256, 0, stream>>>(x, rwse, nW, nb, idxw, rW, rb, h);

    for (int l = 0; l < 3; ++l) {
        fill_kernel<<<(N_NODES * DM + 255) / 256, 256, 0, stream>>>(aggr, 0.0f, N_NODES * DM);
        msg_kernel<<<(N_EDGES + 31) / 32, 256, 0, stream>>>(h, ei, eattr,
                                                            tmsg + (size_t)l * 2720 * 256,
                                                            msg_b + l * 256, aggr);
        upd_kernel<<<N_NODES / 32, 256, 0, stream>>>(h, aggr,
                                                     tupd + (size_t)l * 2560 * 256,
                                                     upd_b + l * 256,
                                                     cg + l * 256, cb + l * 256,
                                                     ng + l * 256, nb2 + l * 256);
    }

    proj_kernel<<<N_NODES / 32, 256, 0, stream>>>(h, tpw, pb, (float*)d_out);

    int tail = out_size - N_NODES * DM;   // mask (all-true -> 1.0f)
    if (tail > 0)
        fill_kernel<<<(tail + 255) / 256, 256, 0, stream>>>((float*)d_out + N_NODES * DM, 1.0f, tail);
}